// MaskRCNN_50835232915728
// MI455X (gfx1250) — compile-verified
//
#include <hip/hip_runtime.h>
#include <math.h>

// ---------------------------------------------------------------------------
// MaskRCNN RPN head for MI455X (gfx1250, wave32, WMMA + async-to-LDS)
//   conv3x3(512->512)+ReLU  : fp32 WMMA 16x16x4 GEMM  (M=2304,N=512,K=4608)
//                             B tiles staged in LDS via GLOBAL_LOAD_ASYNC_TO_LDS
//   1x1 cls/reg + box decode: per-anchor wave dot products
//   top-k 12000 (ordered)   : 32K-key bitonic sort in 256KB dynamic LDS
//   greedy NMS (0.7)        : parallel bitmask + single-wave serial scan
// ---------------------------------------------------------------------------

typedef float v2f __attribute__((ext_vector_type(2)));
typedef float v8f __attribute__((ext_vector_type(8)));

#define HF        48
#define M_TOT     2304          // 48*48 spatial
#define K_TOT     4608          // 512*9
#define C_OUT     512
#define N_ANCH    20736         // 2304*9
#define NPRE      12000
#define NPOST     2000
#define NMS_WORDS 375           // 12000/32
#define SORT_N    32768
#define IMG       768.0f
#define THRESH    0.7f

#define BSTRIDE   34            // padded LDS stride (floats) per n-column

__device__ __forceinline__ unsigned ordf(float f) {
  unsigned u = __float_as_uint(f);
  return (u & 0x80000000u) ? ~u : (u | 0x80000000u);
}

// ------------------------- kernel 0: zero output ---------------------------
__global__ void k_zero(float* __restrict__ out, int n) {
  int i = blockIdx.x * 256 + threadIdx.x;
  if (i < n) out[i] = 0.0f;
}

// ------- kernel 1: reorder weights into per-workgroup-contiguous tiles -----
// Tile (r, chunk, nblk) is 8KB contiguous: [n_local(64)][kk(32)], where
// k = r*512 + chunk*32 + kk and o = nblk*64 + n_local.
__global__ void k_prep_b(const float* __restrict__ w, float* __restrict__ Bprep) {
  int idx = blockIdx.x * 256 + threadIdx.x;       // over 4608*512
  if (idx >= K_TOT * C_OUT) return;
  int tile_id = idx >> 11;                         // / 2048
  int elem    = idx & 2047;
  int n_local = elem >> 5;
  int kk      = elem & 31;
  int nblk    = tile_id & 7;
  int rc      = tile_id >> 3;
  int chunk   = rc & 15;
  int r       = rc >> 4;
  int o = nblk * 64 + n_local;
  int c = chunk * 32 + kk;
  Bprep[idx] = w[o * K_TOT + c * 9 + r];
}

// ---------------- kernel 2: 3x3 conv + bias + ReLU via f32 WMMA ------------
// grid (8, 18), block 128 (4 waves). Wave tile: M=32, N=64 -> 8 accumulators.
// B tile (32k x 64n, 8KB) staged per workgroup via async-to-LDS (ASYNCcnt).
__global__ void __launch_bounds__(128) k_conv_wmma(
    const float* __restrict__ F,      // (512, 48, 48)
    const float* __restrict__ Bprep,  // tiled weights (see k_prep_b)
    const float* __restrict__ bias,   // (512,)
    float* __restrict__ H)            // out (2304, 512)
{
  __shared__ v2f Blds2[64 * (BSTRIDE / 2)];        // 64 n-cols * 34 floats = 8.5KB

  const int tid   = threadIdx.x;
  const int lane  = tid & 31;
  const int wav   = tid >> 5;
  const int half  = lane >> 4;        // 0: K=k0/k0+1 , 1: K=k0+2/k0+3
  const int lr    = lane & 15;
  const int nblk  = blockIdx.x;
  const int n0    = nblk * 64;
  const int mbase = blockIdx.y * 128 + wav * 32;

  float* const BldsF = (float*)Blds2;
  const unsigned ldsbase = (unsigned)(uintptr_t)BldsF;  // low 32 bits = LDS offset
  // async-stage addressing: thread handles 4 consecutive floats of the flat
  // [n][kk] tile per pass -> n_local = e4>>3, kk = (e4&7)*4 (4 | 32, no n-cross)
  const int e4      = tid;                              // pass 0 element/4 index
  const int st_n    = e4 >> 3;
  const int st_kk   = (e4 & 7) * 4;

  v8f acc[2][4];
  const v8f vzero = {0.f,0.f,0.f,0.f,0.f,0.f,0.f,0.f};
#pragma unroll
  for (int s = 0; s < 2; ++s)
#pragma unroll
    for (int t = 0; t < 4; ++t) acc[s][t] = vzero;

  int ym[2], xm[2];
#pragma unroll
  for (int s = 0; s < 2; ++s) {
    int m = mbase + s * 16 + lr;      // A rows: lanes 0-15 / 16-31 both M=0..15
    ym[s] = m / HF;
    xm[s] = m % HF;
  }

  for (int r = 0; r < 9; ++r) {       // tap index (kh*3+kw)
    const int kh = r / 3 - 1, kw = r % 3 - 1;
    int  off[2];
    bool val[2];
#pragma unroll
    for (int s = 0; s < 2; ++s) {
      int yy = ym[s] + kh, xx = xm[s] + kw;
      val[s] = (yy >= 0) & (yy < HF) & (xx >= 0) & (xx < HF);
      off[s] = yy * HF + xx;
    }

    for (int chunk = 0; chunk < 16; ++chunk) {
      const float* __restrict__ tb =
          Bprep + ((size_t)(((r * 16 + chunk) * 8) + nblk) << 11);

      __syncthreads();    // previous chunk fully consumed before overwrite
#pragma unroll
      for (int q = 0; q < 4; ++q) {
        // pass q covers elements [q*512, q*512+512): 128 lanes * 4 floats
        const int nl = st_n + q * 16;                   // 128/8 = 16 n per pass
        const unsigned dst = ldsbase + (unsigned)(nl * BSTRIDE + st_kk) * 4u;
        const float* src = tb + (q * 128 + e4) * 4;
        asm volatile("global_load_async_to_lds_b128 %0, %1, off"
                     :: "v"(dst), "v"(src) : "memory");
      }
      asm volatile("s_wait_asynccnt 0x0" ::: "memory");
      __syncthreads();    // B tile visible to all waves

      const int cbase = chunk * 32;
#pragma unroll
      for (int kk = 0; kk < 32; kk += 4) {
        const int ca  = cbase + kk + 2 * half;          // global k (A side)
        const int kkl = kk + 2 * half;                  // LDS k (B side)
        v2f a[2];
#pragma unroll
        for (int s = 0; s < 2; ++s) {
          a[s].x = val[s] ? F[ca * M_TOT + off[s]]       : 0.f;
          a[s].y = val[s] ? F[(ca + 1) * M_TOT + off[s]] : 0.f;
        }
#pragma unroll
        for (int t = 0; t < 4; ++t) {
          const v2f b = Blds2[(t * 16 + lr) * (BSTRIDE / 2) + (kkl >> 1)];
#pragma unroll
          for (int s = 0; s < 2; ++s)
            acc[s][t] = __builtin_amdgcn_wmma_f32_16x16x4_f32(
                false, a[s], false, b, (short)0, acc[s][t], false, false);
        }
      }
    }
  }

  // C/D layout: VGPR v -> lanes 0-15: M=v, lanes 16-31: M=v+8, N = lane&15
#pragma unroll
  for (int s = 0; s < 2; ++s)
#pragma unroll
    for (int t = 0; t < 4; ++t) {
      const int nn = n0 + t * 16 + lr;
      const float bv = bias[nn];
#pragma unroll
      for (int v = 0; v < 8; ++v) {
        const int mm = mbase + s * 16 + v + 8 * half;
        float r2 = acc[s][t][v] + bv;
        H[(size_t)mm * C_OUT + nn] = r2 > 0.f ? r2 : 0.f;
      }
    }
}

// ---------------- kernel 3: 1x1 heads + anchor decode ----------------------
// one wave per anchor: 5 dot products of length 512 (cls[2a+1], reg[4a..4a+3])
__global__ void __launch_bounds__(32) k_heads(
    const float* __restrict__ H,
    const float* __restrict__ wc, const float* __restrict__ bc,
    const float* __restrict__ wr, const float* __restrict__ br,
    float* __restrict__ scores, float* __restrict__ rois)
{
  const int g = blockIdx.x;            // anchor index
  const int lane = threadIdx.x;
  const int p = g / 9, a = g % 9;
  const float* __restrict__ hrow = H + (size_t)p * C_OUT;
  const float* __restrict__ w0 = wc + (2 * a + 1) * C_OUT;
  const float* __restrict__ w1 = wr + (4 * a + 0) * C_OUT;
  const float* __restrict__ w2 = wr + (4 * a + 1) * C_OUT;
  const float* __restrict__ w3 = wr + (4 * a + 2) * C_OUT;
  const float* __restrict__ w4 = wr + (4 * a + 3) * C_OUT;

  float a0 = 0.f, a1 = 0.f, a2 = 0.f, a3 = 0.f, a4 = 0.f;
  for (int c = lane; c < C_OUT; c += 32) {
    float hv = hrow[c];
    a0 += hv * w0[c]; a1 += hv * w1[c]; a2 += hv * w2[c];
    a3 += hv * w3[c]; a4 += hv * w4[c];
  }
#pragma unroll
  for (int o = 16; o > 0; o >>= 1) {
    a0 += __shfl_down(a0, o, 32);
    a1 += __shfl_down(a1, o, 32);
    a2 += __shfl_down(a2, o, 32);
    a3 += __shfl_down(a3, o, 32);
    a4 += __shfl_down(a4, o, 32);
  }
  if (lane == 0) {
    const float s1 = a0 + bc[2 * a + 1];
    const float dy = a1 + br[4 * a + 0];
    const float dx = a2 + br[4 * a + 1];
    const float dh = a3 + br[4 * a + 2];
    const float dw = a4 + br[4 * a + 3];
    const int i = p / HF, j = p % HF;
    const float cy = 7.5f + 16.f * (float)i;
    const float cx = 7.5f + 16.f * (float)j;
    const float areas[3] = {2000.f, 1000.f, 2000.f / 9.f};
    const float rats[3]  = {0.5f, 1.f, 2.f};
    const float ar = areas[a % 3], rt = rats[a / 3];
    const float hh = 2.f * sqrtf(ar / rt);
    const float ww = 2.f * sqrtf(ar * rt);
    const float dh2 = 0.5f * expf(dh);
    const float dw2 = 0.5f * expf(dw);
    scores[g] = s1;
    rois[g * 4 + 0] = fminf(fmaxf(cy + hh * (dy - dh2), 0.f), IMG);
    rois[g * 4 + 1] = fminf(fmaxf(cx + ww * (dx - dw2), 0.f), IMG);
    rois[g * 4 + 2] = fminf(fmaxf(cy + hh * (dy + dh2), 0.f), IMG);
    rois[g * 4 + 3] = fminf(fmaxf(cx + ww * (dx + dw2), 0.f), IMG);
  }
}

// ---------------- kernel 4: descending bitonic sort in 256KB LDS -----------
// key = (ordf(score) << 32) | (~idx)  -> descending score, ties -> lower idx.
// Writes the 12000 highest-score boxes, in order, to roi_sorted.
__global__ void __launch_bounds__(1024) k_sort(
    const float* __restrict__ scores, const float* __restrict__ rois,
    float* __restrict__ roi_sorted)
{
  extern __shared__ unsigned long long keys[];   // 32768 * 8B = 256 KB LDS
  const int tid = threadIdx.x;
  for (int e = tid; e < SORT_N; e += 1024) {
    if (e < N_ANCH)
      keys[e] = ((unsigned long long)ordf(scores[e]) << 32) |
                (unsigned long long)(0xFFFFFFFFu - (unsigned)e);
    else
      keys[e] = 0ull;                            // smaller than any real key
  }
  __syncthreads();
  for (int k = 2; k <= SORT_N; k <<= 1) {
    for (int j = k >> 1; j > 0; j >>= 1) {
      for (int e = tid; e < SORT_N; e += 1024) {
        const int ix = e ^ j;
        if (ix > e) {
          const unsigned long long va = keys[e], vb = keys[ix];
          const bool desc = ((e & k) == 0);
          if ((va < vb) == desc) { keys[e] = vb; keys[ix] = va; }
        }
      }
      __syncthreads();
    }
  }
  for (int t = tid; t < NPRE; t += 1024) {
    const unsigned idx = 0xFFFFFFFFu - (unsigned)(keys[t] & 0xFFFFFFFFull);
#pragma unroll
    for (int q = 0; q < 4; ++q)
      roi_sorted[t * 4 + q] = rois[(size_t)idx * 4 + q];
  }
}

// ---------------- kernel 5: NMS suppression bitmask ------------------------
// mask[i][w] bit b set <=> j = w*32+b, j > i, and box i suppresses box j.
__global__ void __launch_bounds__(32) k_mask(
    const float* __restrict__ B, unsigned* __restrict__ mask)
{
  __shared__ float jb[32][4];
  const int t  = threadIdx.x;
  const int j0 = blockIdx.x * 32;
  const int i  = blockIdx.y * 32 + t;            // 375*32 == 12000 exactly
#pragma unroll
  for (int q = 0; q < 4; ++q) jb[t][q] = B[(size_t)(j0 + t) * 4 + q];
  __syncthreads();

  const float b0 = B[i * 4 + 0], b1 = B[i * 4 + 1];
  const float b2 = B[i * 4 + 2], b3 = B[i * 4 + 3];
  const float areai = (b2 - b0) * (b3 - b1);
  unsigned wm = 0u;
#pragma unroll 4
  for (int jj = 0; jj < 32; ++jj) {
    const int j = j0 + jj;
    if (j > i) {
      const float c0 = jb[jj][0], c1 = jb[jj][1];
      const float c2 = jb[jj][2], c3 = jb[jj][3];
      const float tl0 = fmaxf(b0, c0), tl1 = fmaxf(b1, c1);
      const float br0 = fminf(b2, c2), br1 = fminf(b3, c3);
      float inter = (br0 - tl0) * (br1 - tl1);
      inter = (tl0 < br0 && tl1 < br1) ? inter : 0.f;
      const float areaj = (c2 - c0) * (c3 - c1);
      const float iou = inter / (areai + areaj - inter);
      if (!(iou < THRESH)) wm |= (1u << jj);     // NaN => suppressed (matches ref)
    }
  }
  mask[(size_t)i * NMS_WORDS + blockIdx.x] = wm;
}

// ---------------- kernel 6: serial greedy scan + emit ----------------------
__global__ void __launch_bounds__(32) k_scan(
    const float* __restrict__ B, const unsigned* __restrict__ mask,
    float* __restrict__ out)
{
  __shared__ unsigned rem[NMS_WORDS];
  const int lane = threadIdx.x;
  for (int w = lane; w < NMS_WORDS; w += 32) rem[w] = 0u;
  __syncthreads();
  int count = 0;
  for (int i = 0; i < NPRE; ++i) {
    const bool kept = ((rem[i >> 5] >> (i & 31)) & 1u) == 0u;   // uniform
    if (kept) {
      const unsigned* __restrict__ mrow = mask + (size_t)i * NMS_WORDS;
      for (int w = lane; w < NMS_WORDS; w += 32) rem[w] |= mrow[w];
      if (count < NPOST && lane < 4) out[count * 4 + lane] = B[i * 4 + lane];
      ++count;
    }
    __syncthreads();
  }
}

// ---------------------------------------------------------------------------
extern "C" void kernel_launch(void* const* d_in, const int* in_sizes, int n_in,
                              void* d_out, int out_size, void* d_ws, size_t ws_size,
                              hipStream_t stream) {
  const float* F     = (const float*)d_in[0];   // (1,512,48,48)
  const float* w_int = (const float*)d_in[1];   // (512,512,3,3)
  const float* b_int = (const float*)d_in[2];   // (512,)
  const float* w_cls = (const float*)d_in[3];   // (18,512,1,1)
  const float* b_cls = (const float*)d_in[4];   // (18,)
  const float* w_reg = (const float*)d_in[5];   // (36,512,1,1)
  const float* b_reg = (const float*)d_in[6];   // (36,)
  float* out = (float*)d_out;                   // (2000,4)

  char* ws = (char*)d_ws;
  size_t off = 0;
  auto alloc = [&](size_t bytes) -> char* {
    char* p = ws + off;
    off += (bytes + 255) & ~(size_t)255;
    return p;
  };
  float*    Bprep  = (float*)alloc((size_t)K_TOT * C_OUT * 4);      // 9.4 MB
  float*    H      = (float*)alloc((size_t)M_TOT * C_OUT * 4);      // 4.7 MB
  float*    scores = (float*)alloc((size_t)N_ANCH * 4);
  float*    rois   = (float*)alloc((size_t)N_ANCH * 16);
  float*    rs     = (float*)alloc((size_t)NPRE * 16);
  unsigned* mask   = (unsigned*)alloc((size_t)NPRE * NMS_WORDS * 4); // 18 MB

  k_zero<<<(NPOST * 4 + 255) / 256, 256, 0, stream>>>(out, NPOST * 4);
  k_prep_b<<<(K_TOT * C_OUT + 255) / 256, 256, 0, stream>>>(w_int, Bprep);
  k_conv_wmma<<<dim3(8, 18), 128, 0, stream>>>(F, Bprep, b_int, H);
  k_heads<<<N_ANCH, 32, 0, stream>>>(H, w_cls, b_cls, w_reg, b_reg, scores, rois);
  k_sort<<<1, 1024, SORT_N * sizeof(unsigned long long), stream>>>(scores, rois, rs);
  k_mask<<<dim3(NMS_WORDS, NMS_WORDS), 32, 0, stream>>>(rs, mask);
  k_scan<<<1, 32, 0, stream>>>(rs, mask, out);
}